// GCLSTM_Model_8581344657591
// MI455X (gfx1250) — compile-verified
//
#include <hip/hip_runtime.h>
#include <math.h>

// GCLSTM with H=C=None: the Cheb-conv over H is identically zero, so
// edge_index/edge_weight/theta are dead. The model collapses to two fused
// dense layers + gating + a linear head:
//   Z1 = x@Wx1 + (bx1+bconv1);  h1 = sig(zo)*tanh(sig(zi)*tanh(zc))
//   Z2 = h1@Wx2 + (bx2+bconv2); h2 = same gating; out = relu(h2)@Wl + bl
// fp32 WMMA (16x16x4) keeps reference numerics; ~5.9 GFLOP on the matrix
// pipe, bound by the 51.2 MB read of x (~2.3us at 23.3 TB/s HBM).

#define N_NODES 100000
#define F_IN    128
#define H1      50
#define H2      20
#define C1      200   // 4*H1 gate columns
#define C2      80    // 4*H2 gate columns

#define NT1     13    // layer-1 column tiles (208 >= 200)
#define NK1     (F_IN / 4)   // 32 K-chunks
#define NT2     5     // layer-2 column tiles (80)
#define NK2     13    // layer-2 K-chunks (52 >= 50)

#define WAVES_PER_BLOCK 4
#define ROWS_PER_BLOCK  (WAVES_PER_BLOCK * 16)

#define Z1_STRIDE 210   // floats; 8*210 % 64 = 16 -> lane-halves hit disjoint banks
#define H1K       52    // layer-2 K padded 50 -> 52 (multiple of 4)
#define Z2_STRIDE 84    // 8*84 % 64 = 32 -> no half-conflict
#define H2S       20

#define B1_PACK_ELEMS (NT1 * NK1 * 32)          // 13312 v2f fragments
#define B1_PACK_BYTES (B1_PACK_ELEMS * 8)       // 106496 bytes of d_ws

typedef __attribute__((ext_vector_type(2))) float v2f;
typedef __attribute__((ext_vector_type(8))) float v8f;

__device__ __forceinline__ float sigmoidf_(float v) {
    return 1.0f / (1.0f + __expf(-v));
}

// ---- pre-pass: repack Wx1 [128x200] into WMMA-B fragment order ----
// P[((t*NK1 + kk)*32 + lane)] = { B[4kk+2h][16t+m], B[4kk+2h+1][16t+m] },
// zero for pad columns (t=12, m>=8).  One-time ~100KB; hot loop then does
// a single coalesced global_load_b64 per fragment with no conditionals.
__global__ __launch_bounds__(128) void repack_wx1_kernel(
    const float* __restrict__ Wx1, v2f* __restrict__ pack)
{
    const int i = blockIdx.x * 128 + threadIdx.x;
    if (i >= B1_PACK_ELEMS) return;
    const int lane = i & 31;
    const int kk   = (i >> 5) % NK1;
    const int t    = (i >> 5) / NK1;
    const int m    = lane & 15;
    const int half = lane >> 4;
    const int krow = kk * 4 + 2 * half;
    const int col  = t * 16 + m;
    v2f b;
    b.x = (col < C1) ? Wx1[krow * C1 + col]       : 0.0f;
    b.y = (col < C1) ? Wx1[(krow + 1) * C1 + col] : 0.0f;
    pack[i] = b;
}

template <bool USE_WS>
__global__ __launch_bounds__(128) void gclstm_fused_kernel(
    const float* __restrict__ x,
    const float* __restrict__ Wx1,                     // direct path only
    const v2f*   __restrict__ Wx1pack,                 // packed path only
    const float* __restrict__ bx1, const float* __restrict__ bconv1,
    const float* __restrict__ Wx2, const float* __restrict__ bx2,
    const float* __restrict__ bconv2,
    const float* __restrict__ Wl,  const float* __restrict__ bl,
    float* __restrict__ out)
{
    __shared__ float sWx2p[NK2 * NT2 * 32 * 2];               // Wx2 in fragment order (16.6KB)
    __shared__ float sBias1[C1];                              // bx1 + bconv1 folded
    __shared__ float sBias2[C2];                              // bx2 + bconv2 folded
    __shared__ float sWl[H2 + 1];                             // Wl, then bl
    __shared__ float sZ1[WAVES_PER_BLOCK][16 * Z1_STRIDE];
    __shared__ float sH1[WAVES_PER_BLOCK][16 * H1K];
    __shared__ float sZ2[WAVES_PER_BLOCK][16 * Z2_STRIDE];
    __shared__ float sH2[WAVES_PER_BLOCK][16 * H2S];

    const int tid  = threadIdx.x;
    const int wave = tid >> 5;
    const int lane = tid & 31;
    const int m    = lane & 15;   // row (A) / column (B, C/D) index within tile
    const int half = lane >> 4;   // selects K pair of the chunk / C row offset

    // ---- stage Wx2 fragments (K rows 50..51 zero), biases, head weights ----
    for (int i = tid; i < NK2 * NT2 * 32; i += 128) {
        const int l2 = i & 31;
        const int t2 = (i >> 5) % NT2;
        const int k2 = (i >> 5) / NT2;
        const int kr = k2 * 4 + 2 * (l2 >> 4);
        const int cc = t2 * 16 + (l2 & 15);
        sWx2p[2 * i]     = (kr < H1)     ? Wx2[kr * C2 + cc]       : 0.0f;
        sWx2p[2 * i + 1] = (kr + 1 < H1) ? Wx2[(kr + 1) * C2 + cc] : 0.0f;
    }
    for (int i = tid; i < C1; i += 128) sBias1[i] = bx1[i] + bconv1[i];
    for (int i = tid; i < C2; i += 128) sBias2[i] = bx2[i] + bconv2[i];
    if (tid < H2) sWl[tid] = Wl[tid];
    if (tid == H2) sWl[H2] = bl[0];
    __syncthreads();

    const int   row0 = (blockIdx.x * WAVES_PER_BLOCK + wave) * 16;
    const int   rowm = min(row0 + m, N_NODES - 1);   // clamp: EXEC stays all-1s for WMMA
    const float* xr  = x + (long)rowm * F_IN;

    // ---- preload layer-1 A fragments: lane {half,m} holds x[rowm][4k+2h .. +1] ----
    v2f afrag[NK1];
#pragma unroll
    for (int kk = 0; kk < NK1; ++kk)
        afrag[kk] = *(const v2f*)(xr + kk * 4 + 2 * half);

    // ---- layer 1: Z1[16x200(+pad)] = Xtile @ Wx1, 13 col-tiles x 32 K WMMAs ----
    float* z1 = &sZ1[wave][0];
#pragma unroll 1
    for (int t = 0; t < NT1; ++t) {
        v8f acc = {};
        if (USE_WS) {
            const v2f* bp = Wx1pack + (t * NK1) * 32 + lane;   // coalesced b64 stream
#pragma unroll
            for (int kk = 0; kk < NK1; ++kk) {
                const v2f b = bp[kk * 32];
                acc = __builtin_amdgcn_wmma_f32_16x16x4_f32(
                    false, afrag[kk], false, b, (short)0, acc, false, false);
            }
        } else {
            const int col = min(t * 16 + m, C1 - 1);           // branch-free pad clamp
#pragma unroll
            for (int kk = 0; kk < NK1; ++kk) {
                const int krow = kk * 4 + 2 * half;
                v2f b;
                b.x = Wx1[krow * C1 + col];
                b.y = Wx1[(krow + 1) * C1 + col];
                acc = __builtin_amdgcn_wmma_f32_16x16x4_f32(
                    false, afrag[kk], false, b, (short)0, acc, false, false);
            }
        }
        // C/D layout: VGPR r: lanes0-15 -> (row r, col lane); lanes16-31 -> (row r+8, col lane-16)
#pragma unroll
        for (int r = 0; r < 8; ++r)
            z1[(r + 8 * half) * Z1_STRIDE + t * 16 + m] = acc[r];
    }

    // ---- layer-1 gates (forget gate dead: C=0); folded biases from LDS ----
    float* h1 = &sH1[wave][0];
    {   // zero the K-pad columns 50,51 for all 16 rows (32 slots, one per lane)
        const int r = lane >> 1, c = H1 + (lane & 1);
        h1[r * H1K + c] = 0.0f;
    }
    for (int e = lane; e < 16 * H1; e += 32) {
        const int r = e / H1, j = e % H1;
        const float zi = z1[r * Z1_STRIDE + j]          + sBias1[j];
        const float zc = z1[r * Z1_STRIDE + j + 2 * H1] + sBias1[j + 2 * H1];
        const float zo = z1[r * Z1_STRIDE + j + 3 * H1] + sBias1[j + 3 * H1];
        h1[r * H1K + j] = sigmoidf_(zo) * tanhf(sigmoidf_(zi) * tanhf(zc));
    }

    // ---- layer 2: Z2[16x80] = h1 @ Wx2 (K padded to 52), fragments from LDS ----
    v8f acc2[NT2] = {};
#pragma unroll 1
    for (int kk = 0; kk < NK2; ++kk) {
        const int krow = kk * 4 + 2 * half;
        const v2f a = *(const v2f*)&h1[m * H1K + krow];        // 8B-aligned (krow even)
#pragma unroll
        for (int t = 0; t < NT2; ++t) {
            const v2f b = *(const v2f*)&sWx2p[2 * ((kk * NT2 + t) * 32 + lane)];
            acc2[t] = __builtin_amdgcn_wmma_f32_16x16x4_f32(
                false, a, false, b, (short)0, acc2[t], false, false);
        }
    }
    float* z2 = &sZ2[wave][0];
#pragma unroll
    for (int t = 0; t < NT2; ++t)
#pragma unroll
        for (int r = 0; r < 8; ++r)
            z2[(r + 8 * half) * Z2_STRIDE + t * 16 + m] = acc2[t][r];

    // ---- layer-2 gates + relu ----
    float* h2 = &sH2[wave][0];
    for (int e = lane; e < 16 * H2; e += 32) {
        const int r = e / H2, j = e % H2;
        const float zi = z2[r * Z2_STRIDE + j]          + sBias2[j];
        const float zc = z2[r * Z2_STRIDE + j + 2 * H2] + sBias2[j + 2 * H2];
        const float zo = z2[r * Z2_STRIDE + j + 3 * H2] + sBias2[j + 3 * H2];
        h2[r * H2S + j] = fmaxf(sigmoidf_(zo) * tanhf(sigmoidf_(zi) * tanhf(zc)), 0.0f);
    }

    // ---- head: out[row] = relu(h2) . Wl + bl (lanes 0..15, one row each) ----
    if (half == 0) {
        const int row = row0 + m;
        if (row < N_NODES) {
            float s = sWl[H2];
#pragma unroll
            for (int j = 0; j < H2; ++j)
                s = fmaf(h2[m * H2S + j], sWl[j], s);
            out[row] = s;
        }
    }
}

extern "C" void kernel_launch(void* const* d_in, const int* in_sizes, int n_in,
                              void* d_out, int out_size, void* d_ws, size_t ws_size,
                              hipStream_t stream) {
    (void)in_sizes; (void)n_in; (void)out_size;
    // setup_inputs() order:
    // 0:x 1:edge_index 2:edge_weight 3:Wx1 4:bx1 5:theta1 6:bconv1
    // 7:Wx2 8:bx2 9:theta2 10:bconv2 11:Wl 12:bl
    const float* x      = (const float*)d_in[0];
    const float* Wx1    = (const float*)d_in[3];
    const float* bx1    = (const float*)d_in[4];
    const float* bconv1 = (const float*)d_in[6];
    const float* Wx2    = (const float*)d_in[7];
    const float* bx2    = (const float*)d_in[8];
    const float* bconv2 = (const float*)d_in[10];
    const float* Wl     = (const float*)d_in[11];
    const float* bl     = (const float*)d_in[12];
    float* out = (float*)d_out;

    const int grid = (N_NODES + ROWS_PER_BLOCK - 1) / ROWS_PER_BLOCK;  // 1563
    const bool use_ws = (d_ws != nullptr) && (ws_size >= (size_t)B1_PACK_BYTES);

    if (use_ws) {
        v2f* pack = (v2f*)d_ws;
        repack_wx1_kernel<<<(B1_PACK_ELEMS + 127) / 128, 128, 0, stream>>>(Wx1, pack);
        gclstm_fused_kernel<true><<<grid, 128, 0, stream>>>(
            x, Wx1, pack, bx1, bconv1, Wx2, bx2, bconv2, Wl, bl, out);
    } else {
        gclstm_fused_kernel<false><<<grid, 128, 0, stream>>>(
            x, Wx1, nullptr, bx1, bconv1, Wx2, bx2, bconv2, Wl, bl, out);
    }
}